// RFAGCN_Drug_78769700208722
// MI455X (gfx1250) — compile-verified
//
#include <hip/hip_runtime.h>
#include <math.h>

#define BB 16
#define NN 2048
#define HH 64
#define TOPK 15
#define ALPHA 0.3f

typedef float v2f __attribute__((ext_vector_type(2)));
typedef float v8f __attribute__((ext_vector_type(8)));

// ---------------------------------------------------------------------------
// 1) Gene embedding: x = relu([ctl,drug] @ W_ge + b_ge), fused imp = x @ attn
//    One wave per node; lane handles channels (lane, lane+32).
// ---------------------------------------------------------------------------
__global__ void embed_kernel(const float* __restrict__ ctl, const float* __restrict__ drug,
                             const float* __restrict__ Wge, const float* __restrict__ bge,
                             const float* __restrict__ attn,
                             float* __restrict__ x, float* __restrict__ imp) {
    int gw = (blockIdx.x * blockDim.x + threadIdx.x) >> 5;
    int lane = threadIdx.x & 31;
    if (gw >= BB * NN) return;
    float c = ctl[gw], d = drug[gw];
    int h0 = lane, h1 = lane + 32;
    float v0 = fmaxf(fmaf(c, Wge[h0], fmaf(d, Wge[HH + h0], bge[h0])), 0.f);
    float v1 = fmaxf(fmaf(c, Wge[h1], fmaf(d, Wge[HH + h1], bge[h1])), 0.f);
    long base = (long)gw * HH;
    x[base + h0] = v0;
    x[base + h1] = v1;
    float s = fmaf(v0, attn[h0], v1 * attn[h1]);
    for (int o = 16; o > 0; o >>= 1) s += __shfl_xor(s, o, 32);
    if (lane == 0) imp[gw] = s;
}

// ---------------------------------------------------------------------------
// 2) Per-batch bitonic sort (descending) of (imp, index) in LDS.
//    One block (1024 threads) per batch; 2048 elements.
// ---------------------------------------------------------------------------
__global__ void __launch_bounds__(1024) sort_kernel(const float* __restrict__ imp,
                                                    float* __restrict__ sVal,
                                                    int* __restrict__ sIdx) {
    __shared__ float v[NN];
    __shared__ int ix[NN];
    int b = blockIdx.x, t = threadIdx.x;
    for (int i = t; i < NN; i += 1024) { v[i] = imp[b * NN + i]; ix[i] = i; }
    __syncthreads();
    for (int k = 2; k <= NN; k <<= 1) {
        for (int j = k >> 1; j > 0; j >>= 1) {
            for (int i = t; i < NN; i += 1024) {
                int l = i ^ j;
                if (l > i) {
                    bool down = ((i & k) == 0);  // descending segment
                    bool sw = down ? (v[i] < v[l]) : (v[i] > v[l]);
                    if (sw) {
                        float tv = v[i]; v[i] = v[l]; v[l] = tv;
                        int ti = ix[i]; ix[i] = ix[l]; ix[l] = ti;
                    }
                }
            }
            __syncthreads();
        }
    }
    for (int i = t; i < NN; i += 1024) { sVal[b * NN + i] = v[i]; sIdx[b * NN + i] = ix[i]; }
}

// ---------------------------------------------------------------------------
// 3) WMMA f32 GEMM: C[M x 64] = A[M x 64] @ W[64 x 64]  (+ per-batch col vec, relu)
//    Block = 128 threads = 4 waves; each wave one 16x16 tile; 16 chained
//    V_WMMA_F32_16X16X4_F32 over K=64. LDS padded to 68 cols (bank-conflict free).
// ---------------------------------------------------------------------------
__global__ void __launch_bounds__(128) gemm64_kernel(const float* __restrict__ A,
                                                     const float* __restrict__ W,
                                                     float* __restrict__ C,
                                                     const float* __restrict__ addv,
                                                     int relu) {
    __shared__ float Alds[16][68];
    __shared__ float Wlds[64][68];
    int t = threadIdx.x;
    int rowBase = blockIdx.x * 16;
    for (int i = t; i < 16 * 64; i += 128)
        Alds[i >> 6][i & 63] = A[(long)(rowBase + (i >> 6)) * 64 + (i & 63)];
    for (int i = t; i < 64 * 64; i += 128)
        Wlds[i >> 6][i & 63] = W[i];
    __syncthreads();

    int wave = t >> 5, lane = t & 31;
    int n0 = wave * 16;
    int m = lane & 15;          // A row / B-C col within tile
    int half = lane >> 4;
    int koff = half << 1;       // ISA 16x4 f32 layout: lanes 0-15 -> K{0,1}, 16-31 -> K{2,3}

    v8f c = {};
#pragma unroll
    for (int kb = 0; kb < 64; kb += 4) {
        v2f a, bf;
        a.x  = Alds[m][kb + koff];
        a.y  = Alds[m][kb + koff + 1];
        bf.x = Wlds[kb + koff][n0 + m];
        bf.y = Wlds[kb + koff + 1][n0 + m];
        c = __builtin_amdgcn_wmma_f32_16x16x4_f32(false, a, false, bf, (short)0, c,
                                                  false, false);
    }

    int col = n0 + m;
    float add = 0.f;
    if (addv) add = addv[(rowBase / NN) * 64 + col];
#pragma unroll
    for (int r = 0; r < 8; ++r) {
        int row = rowBase + r + half * 8;   // C layout: lanes 0-15 -> M 0..7, 16-31 -> M 8..15
        float val = c[r] + add;
        if (relu) val = fmaxf(val, 0.f);
        C[(long)row * 64 + col] = val;
    }
}

// ---------------------------------------------------------------------------
// 4) Top-15 selection (ballot scan over sorted imp) + softmax + weighted
//    neighbor gather + residual/relu, fused next-layer importance.
//    One wave per node; selection k lives in lane k's registers.
// ---------------------------------------------------------------------------
__global__ void aggregate_kernel(const float* __restrict__ xin,
                                 const float* __restrict__ sup,
                                 const float* __restrict__ sVal,
                                 const int* __restrict__ sIdx,
                                 const int* __restrict__ adj,
                                 const float* __restrict__ bias,
                                 const float* __restrict__ attn,   // may be null
                                 float* __restrict__ xout,
                                 float* __restrict__ imp) {
    int gw = (blockIdx.x * blockDim.x + threadIdx.x) >> 5;
    int lane = threadIdx.x & 31;
    if (gw >= BB * NN) return;
    int b = gw / NN, n = gw % NN;
    long arow = (long)n * NN;

    float selV = 0.f;
    int selJ = 0;
    int count = 0;
    for (int r0 = 0; r0 < NN && count < TOPK; r0 += 32) {
        int r = r0 + lane;                       // NN % 32 == 0 -> always in range
        int j = sIdx[b * NN + r];
        float v = sVal[b * NN + r];
        bool hit = adj[arow + j] > 0;
        unsigned mask = (unsigned)__ballot(hit);
        while (mask && count < TOPK) {
            int l = __ffs(mask) - 1;
            mask &= mask - 1;
            int jj = __shfl(j, l, 32);
            float vv = __shfl(v, l, 32);
            if (lane == count) { selJ = jj; selV = vv; }
            ++count;
        }
    }

    // softmax over the selected values (descending -> lane 0 holds the max;
    // the imp_i row offset cancels under softmax so it is omitted entirely)
    float vmax = __shfl(selV, 0, 32);
    float e = (lane < count) ? expf(selV - vmax) : 0.f;
    float sum = e;
    for (int o = 16; o > 0; o >>= 1) sum += __shfl_xor(sum, o, 32);
    float w = (sum > 0.f) ? e / sum : 0.f;

    float agg0 = 0.f, agg1 = 0.f;
    for (int k = 0; k < count; ++k) {
        float wk = __shfl(w, k, 32);
        int jk = __shfl(selJ, k, 32);
        const float* s = sup + ((long)b * NN + jk) * 64;
        agg0 = fmaf(wk, s[lane], agg0);
        agg1 = fmaf(wk, s[lane + 32], agg1);
    }

    long base = (long)gw * 64;
    float o0 = fmaxf(fmaf(ALPHA, xin[base + lane],      (1.f - ALPHA) * agg0) + bias[lane],      0.f);
    float o1 = fmaxf(fmaf(ALPHA, xin[base + lane + 32], (1.f - ALPHA) * agg1) + bias[lane + 32], 0.f);
    xout[base + lane] = o0;
    xout[base + lane + 32] = o1;

    if (attn) {
        float s2 = fmaf(o0, attn[lane], o1 * attn[lane + 32]);
        for (int o = 16; o > 0; o >>= 1) s2 += __shfl_xor(s2, o, 32);
        if (lane == 0) imp[gw] = s2;
    }
}

// ---------------------------------------------------------------------------
// 5) Per-batch head constant: cvec[b] = cell_table[cidx[b]] @ W_d1[64:128] + b_d1
// ---------------------------------------------------------------------------
__global__ void cellprep_kernel(const float* __restrict__ table, const int* __restrict__ cidx,
                                const float* __restrict__ Wd1, const float* __restrict__ bd1,
                                float* __restrict__ cvec) {
    int b = blockIdx.x, o = threadIdx.x;  // 64 threads
    int ci = cidx[b];
    float acc = bd1[o];
    for (int h = 0; h < 64; ++h)
        acc = fmaf(table[ci * 64 + h], Wd1[(64 + h) * 64 + o], acc);
    cvec[b * 64 + o] = acc;
}

// ---------------------------------------------------------------------------
// 6) Final: out = ctl + Hh @ W_d2 + b_d2   (wave-per-node dot)
// ---------------------------------------------------------------------------
__global__ void final_kernel(const float* __restrict__ Hh, const float* __restrict__ Wd2,
                             const float* __restrict__ bd2, const float* __restrict__ ctl,
                             float* __restrict__ out) {
    int gw = (blockIdx.x * blockDim.x + threadIdx.x) >> 5;
    int lane = threadIdx.x & 31;
    if (gw >= BB * NN) return;
    long base = (long)gw * 64;
    float s = fmaf(Hh[base + lane], Wd2[lane], Hh[base + lane + 32] * Wd2[lane + 32]);
    for (int o = 16; o > 0; o >>= 1) s += __shfl_xor(s, o, 32);
    if (lane == 0) out[gw] = ctl[gw] + s + bd2[0];
}

// ---------------------------------------------------------------------------
extern "C" void kernel_launch(void* const* d_in, const int* in_sizes, int n_in,
                              void* d_out, int out_size, void* d_ws, size_t ws_size,
                              hipStream_t stream) {
    (void)in_sizes; (void)n_in; (void)out_size; (void)ws_size;
    const int*   adj   = (const int*)d_in[0];
    const float* ctl   = (const float*)d_in[1];
    const float* drug  = (const float*)d_in[2];
    const int*   cidx  = (const int*)d_in[3];
    const float* Wge   = (const float*)d_in[4];
    const float* bge   = (const float*)d_in[5];
    const float* ctab  = (const float*)d_in[6];
    const float* attn1 = (const float*)d_in[7];
    const float* k1    = (const float*)d_in[8];
    const float* b1    = (const float*)d_in[9];
    const float* attn2 = (const float*)d_in[10];
    const float* k2    = (const float*)d_in[11];
    const float* b2    = (const float*)d_in[12];
    const float* Wd1   = (const float*)d_in[13];
    const float* bd1   = (const float*)d_in[14];
    const float* Wd2   = (const float*)d_in[15];
    const float* bd2   = (const float*)d_in[16];
    float* out = (float*)d_out;

    char* ws = (char*)d_ws;
    size_t off = 0;
    auto alloc = [&](size_t bytes) -> void* {
        void* p = ws + off;
        off += (bytes + 255) & ~(size_t)255;
        return p;
    };
    float* x1   = (float*)alloc((size_t)BB * NN * HH * 4);
    float* x2   = (float*)alloc((size_t)BB * NN * HH * 4);
    float* sup  = (float*)alloc((size_t)BB * NN * HH * 4);
    float* imp  = (float*)alloc((size_t)BB * NN * 4);
    float* sVal = (float*)alloc((size_t)BB * NN * 4);
    int*   sIdx = (int*)alloc((size_t)BB * NN * 4);
    float* cvec = (float*)alloc((size_t)BB * 64 * 4);
    float* x3 = x1;   // x1 dead after layer-1 aggregate
    float* Hh = sup;  // sup dead after layer-2 aggregate

    dim3 blk(256);
    dim3 nodeGrid((BB * NN * 32) / 256);   // one wave per node
    int gemmGrid = (BB * NN) / 16;

    // layer 1
    embed_kernel<<<nodeGrid, blk, 0, stream>>>(ctl, drug, Wge, bge, attn1, x1, imp);
    sort_kernel<<<BB, 1024, 0, stream>>>(imp, sVal, sIdx);
    gemm64_kernel<<<gemmGrid, 128, 0, stream>>>(x1, k1, sup, nullptr, 0);
    aggregate_kernel<<<nodeGrid, blk, 0, stream>>>(x1, sup, sVal, sIdx, adj, b1, attn2, x2, imp);
    // layer 2
    sort_kernel<<<BB, 1024, 0, stream>>>(imp, sVal, sIdx);
    gemm64_kernel<<<gemmGrid, 128, 0, stream>>>(x2, k2, sup, nullptr, 0);
    aggregate_kernel<<<nodeGrid, blk, 0, stream>>>(x2, sup, sVal, sIdx, adj, b2, nullptr, x3, imp);
    // head
    cellprep_kernel<<<BB, 64, 0, stream>>>(ctab, cidx, Wd1, bd1, cvec);
    gemm64_kernel<<<gemmGrid, 128, 0, stream>>>(x3, Wd1, Hh, cvec, 1);
    final_kernel<<<nodeGrid, blk, 0, stream>>>(Hh, Wd2, bd2, ctl, out);
}